// SharedMolecularFeatureExtractor_45251775431202
// MI455X (gfx1250) — compile-verified
//
#include <hip/hip_runtime.h>

#define DIM    128
#define NTYPES 44
#define NFIX   34
#define XCOLS  78     // NTYPES + NFIX
#define KLIN   164    // 128 + 34 = 162, padded to multiple of 4
#define APAD   132    // 128 + 4 LDS padding (bank-conflict avoidance)

typedef __attribute__((ext_vector_type(2))) float v2f;
typedef __attribute__((ext_vector_type(8))) float v8f;

// ---------------------------------------------------------------------------
// Kernel 1: embedding gather + concat + linear [162->128] + ReLU, via
// V_WMMA_F32_16X16X4_F32. One block = 16 nodes; 8 waves = 8 col tiles.
// ---------------------------------------------------------------------------
__global__ __launch_bounds__(256) void k_feat_linear(
    const float* __restrict__ x,     // [N, 78]
    const float* __restrict__ emb,   // [44, 128]
    const float* __restrict__ linW,  // [128, 162] row-major
    const float* __restrict__ linb,  // [128]
    float* __restrict__ h,           // [N, 128]
    int n)
{
  __shared__ float sfeat[16][KLIN];
  __shared__ int   sidx[16];
  const int tid  = threadIdx.x;
  const int base = blockIdx.x * 16;

  // Phase 1: argmax over one-hot atom-type block (one thread per node)
  if (tid < 16) {
    int node = base + tid; if (node >= n) node = n - 1;
    const float* xr = x + (size_t)node * XCOLS;
    float best = xr[0]; int bi = 0;
    for (int c = 1; c < NTYPES; ++c) {
      float v = xr[c];
      if (v > best) { best = v; bi = c; }
    }
    sidx[tid] = bi;
  }
  __syncthreads();

  // Phase 2: assemble feature rows: [ emb[idx] (128) | x[:,44:78] (34) | 0,0 ]
  for (int p = tid; p < 16 * KLIN; p += 256) {
    const int i = p / KLIN, k = p - i * KLIN;
    int node = base + i; if (node >= n) node = n - 1;
    float v;
    if (k < DIM)             v = emb[sidx[i] * DIM + k];
    else if (k < DIM + NFIX) v = x[(size_t)node * XCOLS + NTYPES + (k - DIM)];
    else                     v = 0.f;
    sfeat[i][k] = v;
  }
  __syncthreads();

  // Phase 3: WMMA. out[m][d] = sum_k feat[m][k] * linW[d][k]
  const int lane = tid & 31;
  const int wave = tid >> 5;
  const int m    = lane & 15;           // A row / D column-lane
  const int kk   = (lane >> 4) << 1;    // 0 (lanes 0-15) or 2 (lanes 16-31)
  const int dcol = wave * 16 + m;       // output feature index

  v8f c = {};
  #pragma unroll
  for (int k0 = 0; k0 < KLIN; k0 += 4) {
    v2f a, b;
    const int ka = k0 + kk, kb = k0 + kk + 1;
    a.x = sfeat[m][ka];
    a.y = sfeat[m][kb];
    b.x = (ka < 162) ? linW[dcol * 162 + ka] : 0.f;
    b.y = (kb < 162) ? linW[dcol * 162 + kb] : 0.f;
    c = __builtin_amdgcn_wmma_f32_16x16x4_f32(false, a, false, b,
                                              (short)0, c, false, false);
  }

  // Epilogue: bias + ReLU. D layout: VGPR r -> row r (lanes 0-15) / r+8.
  const float bias  = linb[dcol];
  const int   mbase = (lane >> 4) * 8;
  #pragma unroll
  for (int r = 0; r < 8; ++r) {
    const int row = base + mbase + r;
    if (row < n)
      h[(size_t)row * DIM + dcol] = fmaxf(c[r] + bias, 0.f);
  }
}

// ---------------------------------------------------------------------------
// Kernel 2: GEMM hout = (relu?)(hin) @ W^T, W [128,128] row-major. WMMA f32.
// ---------------------------------------------------------------------------
__global__ __launch_bounds__(256) void k_gemm128(
    const float* __restrict__ hin,  // [N,128]
    const float* __restrict__ W,    // [128,128]
    float* __restrict__ hout,       // [N,128]
    int relu_in, int n)
{
  __shared__ float sA[16][APAD];
  const int tid  = threadIdx.x;
  const int base = blockIdx.x * 16;

  // Cooperative A-tile load (16 x 128, float4 vectorized)
  for (int p = tid; p < 16 * (DIM / 4); p += 256) {
    const int i = p >> 5, k4 = (p & 31) * 4;
    int node = base + i; if (node >= n) node = n - 1;
    float4 v = *(const float4*)(hin + (size_t)node * DIM + k4);
    if (relu_in) {
      v.x = fmaxf(v.x, 0.f); v.y = fmaxf(v.y, 0.f);
      v.z = fmaxf(v.z, 0.f); v.w = fmaxf(v.w, 0.f);
    }
    *(float4*)(&sA[i][k4]) = v;
  }
  __syncthreads();

  const int lane = tid & 31;
  const int wave = tid >> 5;
  const int m    = lane & 15;
  const int kk   = (lane >> 4) << 1;
  const int dcol = wave * 16 + m;

  v8f c = {};
  #pragma unroll
  for (int k0 = 0; k0 < DIM; k0 += 4) {
    v2f a, b;
    a.x = sA[m][k0 + kk];
    a.y = sA[m][k0 + kk + 1];
    b.x = W[dcol * DIM + k0 + kk];
    b.y = W[dcol * DIM + k0 + kk + 1];
    c = __builtin_amdgcn_wmma_f32_16x16x4_f32(false, a, false, b,
                                              (short)0, c, false, false);
  }

  const int mbase = (lane >> 4) * 8;
  #pragma unroll
  for (int r = 0; r < 8; ++r) {
    const int row = base + mbase + r;
    if (row < n)
      hout[(size_t)row * DIM + dcol] = c[r];
  }
}

// ---------------------------------------------------------------------------
// Degree / normalization kernels
// ---------------------------------------------------------------------------
__global__ void k_deg_init(float* __restrict__ deg, int n) {
  const int i = blockIdx.x * 256 + threadIdx.x;
  if (i < n) deg[i] = 1.f;  // self-loop contribution
}
__global__ void k_deg_accum(const int* __restrict__ dst, float* __restrict__ deg, int e) {
  const int i = blockIdx.x * 256 + threadIdx.x;
  if (i < e) atomicAdd(&deg[dst[i]], 1.f);
}
__global__ void k_dinv(const float* __restrict__ deg, float* __restrict__ dinv, int n) {
  const int i = blockIdx.x * 256 + threadIdx.x;
  if (i < n) dinv[i] = rsqrtf(deg[i]);  // deg >= 1 always (self-loops)
}

// ---------------------------------------------------------------------------
// Scatter: init with bias + self-loop message, then edge atomics, then ReLU.
// ---------------------------------------------------------------------------
__global__ void k_self_init(const float* __restrict__ tmp,
                            const float* __restrict__ dinv,
                            const float* __restrict__ bias,
                            float* __restrict__ out, int n)
{
  const int i = blockIdx.x * 256 + threadIdx.x;
  if (i < n * DIM) {
    const int node = i >> 7, d = i & (DIM - 1);
    const float di = dinv[node];
    out[i] = bias[d] + tmp[i] * di * di;
  }
}

__global__ __launch_bounds__(256) void k_edge_scatter(
    const int* __restrict__ src, const int* __restrict__ dst,
    const float* __restrict__ dinv, const float* __restrict__ tmp,
    float* __restrict__ out, int e)
{
  const int warp = (blockIdx.x * 256 + threadIdx.x) >> 5;  // one wave32 / edge
  const int lane = threadIdx.x & 31;
  if (warp >= e) return;
  const int s = src[warp], d = dst[warp];
  const float w = dinv[s] * dinv[d];
  const float4 v = *(const float4*)(tmp + (size_t)s * DIM + lane * 4);
  float* o = out + (size_t)d * DIM + lane * 4;
  atomicAdd(o + 0, v.x * w);
  atomicAdd(o + 1, v.y * w);
  atomicAdd(o + 2, v.z * w);
  atomicAdd(o + 3, v.w * w);
}

__global__ void k_relu(float* __restrict__ a, int n) {
  const int i = blockIdx.x * 256 + threadIdx.x;
  if (i < n) a[i] = fmaxf(a[i], 0.f);
}

// ---------------------------------------------------------------------------
// Launcher
// ---------------------------------------------------------------------------
extern "C" void kernel_launch(void* const* d_in, const int* in_sizes, int n_in,
                              void* d_out, int out_size, void* d_ws, size_t ws_size,
                              hipStream_t stream)
{
  const float* x    = (const float*)d_in[0];
  const int*   ei   = (const int*)  d_in[1];   // [2,E] int32: row0=src, row1=dst
  const float* emb  = (const float*)d_in[3];
  const float* linW = (const float*)d_in[4];
  const float* linb = (const float*)d_in[5];
  const float* W1   = (const float*)d_in[6];
  const float* b1   = (const float*)d_in[7];
  const float* W2   = (const float*)d_in[8];
  const float* b2   = (const float*)d_in[9];
  float* out = (float*)d_out;

  const int n = in_sizes[0] / XCOLS;
  const int e = in_sizes[1] / 2;
  const int* src = ei;
  const int* dst = ei + e;

  // Workspace layout (floats)
  float* h0   = (float*)d_ws;                    // [N,128]
  float* tmp  = h0   + (size_t)n * DIM;          // [N,128]
  float* acc1 = tmp  + (size_t)n * DIM;          // [N,128]
  float* deg  = acc1 + (size_t)n * DIM;          // [N]
  float* dinv = deg + n;                         // [N]

  const int tiles  = (n + 15) / 16;
  const int nblk   = (n + 255) / 256;
  const int eblk   = (e + 255) / 256;
  const int ndblk  = (n * DIM + 255) / 256;
  const int scblk  = (e + 7) / 8;                // 8 wave32 edges per block

  // 1) features + linear + relu  -> h0
  k_feat_linear<<<tiles, 256, 0, stream>>>(x, emb, linW, linb, h0, n);

  // 2) degree + rsqrt (shared by both layers)
  k_deg_init <<<nblk, 256, 0, stream>>>(deg, n);
  k_deg_accum<<<eblk, 256, 0, stream>>>(dst, deg, e);
  k_dinv     <<<nblk, 256, 0, stream>>>(deg, dinv, n);

  // 3) GCN layer 1: tmp = h0 @ W1^T ; acc1 = b1 + self + edge scatter
  k_gemm128     <<<tiles, 256, 0, stream>>>(h0, W1, tmp, 0, n);
  k_self_init   <<<ndblk, 256, 0, stream>>>(tmp, dinv, b1, acc1, n);
  k_edge_scatter<<<scblk, 256, 0, stream>>>(src, dst, dinv, tmp, acc1, e);

  // 4) GCN layer 2 (ReLU of layer-1 fused into GEMM A-load): -> d_out
  k_gemm128     <<<tiles, 256, 0, stream>>>(acc1, W2, tmp, 1, n);
  k_self_init   <<<ndblk, 256, 0, stream>>>(tmp, dinv, b2, out, n);
  k_edge_scatter<<<scblk, 256, 0, stream>>>(src, dst, dinv, tmp, out, e);

  // 5) final ReLU
  k_relu<<<ndblk, 256, 0, stream>>>(out, n * DIM);
}